// PolicyNetwork_54863912239472
// MI455X (gfx1250) — compile-verified
//
#include <hip/hip_runtime.h>
#include <hip/hip_bf16.h>
#include <math.h>

#define HDIM 64

typedef float v2f __attribute__((ext_vector_type(2)));
typedef float v8f __attribute__((ext_vector_type(8)));

// ---------------------------------------------------------------- utilities
__global__ void fill_kernel(float* __restrict__ p, float v, long n) {
    long i = (long)blockIdx.x * blockDim.x + threadIdx.x;
    long stride = (long)gridDim.x * blockDim.x;
    for (; i < n; i += stride) p[i] = v;
}

// deg[dst] += ew  (deg pre-initialized to 1.0 == self-loop weight)
__global__ void deg_scatter_kernel(const float* __restrict__ ew,
                                   const int* __restrict__ dst,
                                   float* __restrict__ deg, int E) {
    int e = blockIdx.x * blockDim.x + threadIdx.x;
    if (e < E) atomicAdd(&deg[dst[e]], ew[e]);
}

// in-place deg -> deg^{-1/2}
__global__ void dinv_kernel(float* __restrict__ d, int n) {
    int i = blockIdx.x * blockDim.x + threadIdx.x;
    if (i < n) { float v = d[i]; d[i] = (v > 0.f) ? rsqrtf(v) : 0.f; }
}

// ---------------------------------------------------------------- WMMA GEMM
// Y[nrows,64] = X[nrows,64] @ W[64,64], f32, nrows % 16 == 0.
// One wave = one 16x16 D tile; block of 128 threads = 4 waves = 16x64 panel.
// K-loop fully unrolled: 16 x v_wmma_f32_16x16x4_f32.
__global__ __launch_bounds__(128)
void gemm64_wmma_kernel(const float* __restrict__ X, const float* __restrict__ W,
                        float* __restrict__ Y, int nrows) {
    const int lane    = threadIdx.x & 31;
    const int wave    = threadIdx.x >> 5;        // 0..3 -> N tile
    const int rowBase = blockIdx.x * 16;
    const int m       = lane & 15;               // A row / D col index
    const int koff    = (lane >> 4) * 2;         // lanes 16..31 hold K=2,3
    const int colBase = wave * 16;

    v8f c = {0.f, 0.f, 0.f, 0.f, 0.f, 0.f, 0.f, 0.f};
    const float* xrow = X + (size_t)(rowBase + m) * HDIM;

#pragma unroll
    for (int k = 0; k < HDIM; k += 4) {
        v2f a, b;
        // A 16x4 tile: a[0]=X[m][k+koff], a[1]=X[m][k+koff+1]  (global_load_b64)
        const float2 av = *(const float2*)(xrow + k + koff);
        a[0] = av.x; a[1] = av.y;
        // B 4x16 tile: b[0]=W[k+koff][n], b[1]=W[k+koff+1][n]
        b[0] = W[(size_t)(k + koff)     * HDIM + colBase + m];
        b[1] = W[(size_t)(k + koff + 1) * HDIM + colBase + m];
        c = __builtin_amdgcn_wmma_f32_16x16x4_f32(
                /*neg_a=*/false, a, /*neg_b=*/false, b,
                /*c_mod=*/(short)0, c, /*reuse_a=*/false, /*reuse_b=*/false);
    }
    // D 16x16: VGPR v -> row v (lanes 0..15) or v+8 (lanes 16..31), col = lane&15
    const int rbase = rowBase + ((lane >> 4) << 3);
#pragma unroll
    for (int v = 0; v < 8; ++v)
        Y[(size_t)(rbase + v) * HDIM + colBase + m] = c[v];
}

// ---------------------------------------------------------------- edge pass
// One wave per edge: lane l moves feature elements 2l, 2l+1.
// out[dst] += feat[src] * (dinv[src] * ew * dinv[dst])
__global__ __launch_bounds__(256)
void edge_scatter_kernel(const float* __restrict__ feat,
                         const float* __restrict__ dinv,
                         const int* __restrict__ src, const int* __restrict__ dst,
                         const float* __restrict__ ew,
                         float* __restrict__ out, int E) {
    int wid  = (int)((blockIdx.x * (unsigned)blockDim.x + threadIdx.x) >> 5);
    int lane = threadIdx.x & 31;
    if (wid >= E) return;
    int s = src[wid], d = dst[wid];
    float coeff = dinv[s] * ew[wid] * dinv[d];
    float2 v = *(const float2*)(feat + (size_t)s * HDIM + 2 * lane);
    float* o = out + (size_t)d * HDIM + 2 * lane;
    atomicAdd(o,     v.x * coeff);      // global_atomic_add_f32
    atomicAdd(o + 1, v.y * coeff);
}

// out = relu(agg + lin*dinv^2 (self loop) + bias)
__global__ void bias_relu_kernel(const float* __restrict__ agg,
                                 const float* __restrict__ lin,
                                 const float* __restrict__ dinv,
                                 const float* __restrict__ b,
                                 float* __restrict__ out, long n) {
    long i = (long)blockIdx.x * blockDim.x + threadIdx.x;
    if (i >= n) return;
    int node = (int)(i >> 6);
    int j    = (int)(i & 63);
    float di = dinv[node];
    float v  = agg[i] + lin[i] * di * di + b[j];
    out[i] = fmaxf(v, 0.f);
}

// ---------------------------------------------------------------- heads
// One wave per node: two 64-wide dot products via wave32 shfl-xor reduction.
__global__ __launch_bounds__(256)
void heads_kernel(const float* __restrict__ h,
                  const float* __restrict__ Wn, const float* __restrict__ bn,
                  const float* __restrict__ Wr, const float* __restrict__ br,
                  float* __restrict__ logits, float* __restrict__ rescue, int nnodes) {
    int node = (int)((blockIdx.x * (unsigned)blockDim.x + threadIdx.x) >> 5);
    int lane = threadIdx.x & 31;
    if (node >= nnodes) return;
    float2 v  = *(const float2*)(h  + (size_t)node * HDIM + 2 * lane);
    float2 wn = *(const float2*)(Wn + 2 * lane);
    float2 wr = *(const float2*)(Wr + 2 * lane);
    float pn = v.x * wn.x + v.y * wn.y;
    float pr = v.x * wr.x + v.y * wr.y;
#pragma unroll
    for (int off = 16; off; off >>= 1) {
        pn += __shfl_xor(pn, off, 32);
        pr += __shfl_xor(pr, off, 32);
    }
    if (lane == 0) {
        logits[node] = pn + bn[0];
        rescue[node] = 0.01f / (1.f + expf(-(pr + br[0])));
    }
}

// ---------------------------------------------------------------- softmax (deterministic 2-pass)
__global__ __launch_bounds__(256)
void reduce_max_p1(const float* __restrict__ x, float* __restrict__ part, int n) {
    __shared__ float sm[256];
    float m = -INFINITY;
    for (int i = blockIdx.x * blockDim.x + threadIdx.x; i < n; i += gridDim.x * blockDim.x)
        m = fmaxf(m, x[i]);
    sm[threadIdx.x] = m; __syncthreads();
    for (int s = 128; s; s >>= 1) {
        if (threadIdx.x < s) sm[threadIdx.x] = fmaxf(sm[threadIdx.x], sm[threadIdx.x + s]);
        __syncthreads();
    }
    if (threadIdx.x == 0) part[blockIdx.x] = sm[0];
}

__global__ __launch_bounds__(256)
void reduce_max_p2(const float* __restrict__ part, float* __restrict__ red) {
    __shared__ float sm[256];
    sm[threadIdx.x] = part[threadIdx.x]; __syncthreads();
    for (int s = 128; s; s >>= 1) {
        if (threadIdx.x < s) sm[threadIdx.x] = fmaxf(sm[threadIdx.x], sm[threadIdx.x + s]);
        __syncthreads();
    }
    if (threadIdx.x == 0) red[0] = sm[0];
}

__global__ __launch_bounds__(256)
void reduce_sumexp_p1(const float* __restrict__ x, const float* __restrict__ red,
                      float* __restrict__ part, int n) {
    __shared__ float sm[256];
    float mx = red[0];
    float s = 0.f;
    for (int i = blockIdx.x * blockDim.x + threadIdx.x; i < n; i += gridDim.x * blockDim.x)
        s += expf(x[i] - mx);
    sm[threadIdx.x] = s; __syncthreads();
    for (int t = 128; t; t >>= 1) {
        if (threadIdx.x < t) sm[threadIdx.x] += sm[threadIdx.x + t];
        __syncthreads();
    }
    if (threadIdx.x == 0) part[blockIdx.x] = sm[0];
}

__global__ __launch_bounds__(256)
void reduce_sum_p2(const float* __restrict__ part, float* __restrict__ red) {
    __shared__ float sm[256];
    sm[threadIdx.x] = part[threadIdx.x]; __syncthreads();
    for (int t = 128; t; t >>= 1) {
        if (threadIdx.x < t) sm[threadIdx.x] += sm[threadIdx.x + t];
        __syncthreads();
    }
    if (threadIdx.x == 0) red[1] = sm[0];
}

__global__ void softmax_final_kernel(const float* __restrict__ logits,
                                     const float* __restrict__ red,
                                     float* __restrict__ out, int n) {
    int i = blockIdx.x * blockDim.x + threadIdx.x;
    if (i < n) out[i] = expf(logits[i] - red[0]) / red[1];
}

// ---------------------------------------------------------------- launcher
extern "C" void kernel_launch(void* const* d_in, const int* in_sizes, int n_in,
                              void* d_out, int out_size, void* d_ws, size_t ws_size,
                              hipStream_t stream) {
    const float* x  = (const float*)d_in[0];
    const int*   ei = (const int*)d_in[1];      // [2,E] flat
    const float* ew = (const float*)d_in[2];
    const float* W1 = (const float*)d_in[3];
    const float* b1 = (const float*)d_in[4];
    const float* W2 = (const float*)d_in[5];
    const float* b2 = (const float*)d_in[6];
    const float* Wn = (const float*)d_in[7];
    const float* bn = (const float*)d_in[8];
    const float* Wr = (const float*)d_in[9];
    const float* br = (const float*)d_in[10];

    const int E = in_sizes[2];
    const int N = in_sizes[0] / HDIM;
    const int* src = ei;
    const int* dst = ei + E;

    float* out_sel = (float*)d_out;       // [N] softmax
    float* out_rsc = out_sel + N;         // [N] rescue ratios

    float* bufA   = (float*)d_ws;                      // N*64  (linear output)
    float* bufB   = bufA + (size_t)N * HDIM;           // N*64  (aggregation / h)
    float* dinv   = bufB + (size_t)N * HDIM;           // N     (deg -> dinv)
    float* logits = dinv + N;                          // N
    float* part   = logits + N;                        // 256
    float* red    = part + 256;                        // 2 (max, sum)

    const long NH = (long)N * HDIM;
    const int TB = 256;

    // --- degree / normalization -----------------------------------------
    fill_kernel<<<(N + TB - 1) / TB, TB, 0, stream>>>(dinv, 1.0f, N);   // self loop w=1
    deg_scatter_kernel<<<(E + TB - 1) / TB, TB, 0, stream>>>(ew, dst, dinv, E);
    dinv_kernel<<<(N + TB - 1) / TB, TB, 0, stream>>>(dinv, N);

    // --- layer 1 ---------------------------------------------------------
    gemm64_wmma_kernel<<<N / 16, 128, 0, stream>>>(x, W1, bufA, N);
    fill_kernel<<<4096, TB, 0, stream>>>(bufB, 0.f, NH);
    edge_scatter_kernel<<<(E * 32 + TB - 1) / TB, TB, 0, stream>>>(bufA, dinv, src, dst, ew, bufB, E);
    bias_relu_kernel<<<(int)((NH + TB - 1) / TB), TB, 0, stream>>>(bufB, bufA, dinv, b1, bufB, NH);

    // --- layer 2 ---------------------------------------------------------
    gemm64_wmma_kernel<<<N / 16, 128, 0, stream>>>(bufB, W2, bufA, N);
    fill_kernel<<<4096, TB, 0, stream>>>(bufB, 0.f, NH);
    edge_scatter_kernel<<<(E * 32 + TB - 1) / TB, TB, 0, stream>>>(bufA, dinv, src, dst, ew, bufB, E);
    bias_relu_kernel<<<(int)((NH + TB - 1) / TB), TB, 0, stream>>>(bufB, bufA, dinv, b2, bufB, NH);

    // --- heads + softmax -------------------------------------------------
    heads_kernel<<<(N * 32 + TB - 1) / TB, TB, 0, stream>>>(bufB, Wn, bn, Wr, br, logits, out_rsc, N);
    reduce_max_p1<<<256, TB, 0, stream>>>(logits, part, N);
    reduce_max_p2<<<1, TB, 0, stream>>>(part, red);
    reduce_sumexp_p1<<<256, TB, 0, stream>>>(logits, red, part, N);
    reduce_sum_p2<<<1, TB, 0, stream>>>(part, red);
    softmax_final_kernel<<<(N + TB - 1) / TB, TB, 0, stream>>>(logits, red, out_sel, N);
}